// LDDMMHamilton_43731357008440
// MI455X (gfx1250) — compile-verified
//
#include <hip/hip_runtime.h>

// LDDMM Hamiltonian evolve: out = (-dH/dpos, dH/dmom) for
// H = sum_ij exp(-|x_i-x_j|^2/sigma^2) <m_i, m_j>, sigma = 0.1.
//
// Per 16x16 tile: G = X_i X_j^T and P = M_i M_j^T via V_WMMA_F32_16X16X4_F32
// (K=3 padded to 4).  d2 = xx_i + xx_j - 2G, K = exp2(log2e * -d2/s^2),
// w = P*K.  Per-lane row accumulators, shfl_xor half-wave reduce, LDS combine.

typedef __attribute__((ext_vector_type(2))) float v2f;
typedef __attribute__((ext_vector_type(8))) float v8f;

#define INV_S2   100.0f                     // 1/sigma^2
#define LOG2E    1.4426950408889634f
#define C1       (INV_S2 * LOG2E)           // 144.2695...
#define POSCOEF  400.0f                     // 4/sigma^2

__global__ __launch_bounds__(256)
void LDDMMHamilton_wmma_kernel(const float* __restrict__ mom,
                               const float* __restrict__ pos,
                               float* __restrict__ out, int N) {
  const int tid  = threadIdx.x;
  const int lane = tid & 31;
  const int wave = tid >> 5;
  const int half = lane >> 4;     // which 16-lane half of the wave
  const int col  = lane & 15;     // N-column of C/D tile owned by this lane
  const int i0   = blockIdx.x * 16;
  const int ntiles = N >> 4;

  // per-wave partial sums: [wave][row 0..15][q], q = {Sw, Swx,Swy,Swz, Kmx,Kmy,Kmz}
  __shared__ float red[8][16][7];

  // ---- A operands for this i-tile (x and m), f32 16x4 A-matrix layout:
  // lanes 0-15: v0 = K0(x), v1 = K1(y); lanes 16-31: v0 = K2(z), v1 = K3(=0 pad)
  const int arow = i0 + col;
  float ax0, ax1, am0, am1;
  if (half == 0) {
    ax0 = pos[3 * arow + 0]; ax1 = pos[3 * arow + 1];
    am0 = mom[3 * arow + 0]; am1 = mom[3 * arow + 1];
  } else {
    ax0 = pos[3 * arow + 2]; ax1 = 0.0f;
    am0 = mom[3 * arow + 2]; am1 = 0.0f;
  }
  v2f ax = {ax0, ax1};
  v2f am = {am0, am1};

  // -C1 * |x_i|^2 for the 8 rows this lane owns in the C/D layout
  float negrow[8];
#pragma unroll
  for (int r = 0; r < 8; ++r) {
    int row = i0 + r + 8 * half;
    float px = pos[3 * row + 0], py = pos[3 * row + 1], pz = pos[3 * row + 2];
    negrow[r] = -C1 * (px * px + py * py + pz * pz);
  }

  float aKx[8], aKy[8], aKz[8], aW[8], aWx[8], aWy[8], aWz[8];
#pragma unroll
  for (int r = 0; r < 8; ++r) {
    aKx[r] = 0.0f; aKy[r] = 0.0f; aKz[r] = 0.0f; aW[r] = 0.0f;
    aWx[r] = 0.0f; aWy[r] = 0.0f; aWz[r] = 0.0f;
  }

  const v8f czero = {0.0f, 0.0f, 0.0f, 0.0f, 0.0f, 0.0f, 0.0f, 0.0f};

  // ---- sweep j tiles, 8 waves interleaved
  for (int jt = wave; jt < ntiles; jt += 8) {
    const int jrow = jt * 16 + col;
    const float xjx = pos[3 * jrow + 0];
    const float xjy = pos[3 * jrow + 1];
    const float xjz = pos[3 * jrow + 2];
    const float mjx = mom[3 * jrow + 0];
    const float mjy = mom[3 * jrow + 1];
    const float mjz = mom[3 * jrow + 2];

    // B = X_j^T / M_j^T in f32 4x16 B layout (same gather pattern as A layout)
    v2f bx = {half ? xjz : xjx, half ? 0.0f : xjy};
    v2f bm = {half ? mjz : mjx, half ? 0.0f : mjy};

    // G[m][n] = x_i[m] . x_j[n] ; P[m][n] = m_i[m] . m_j[n]
    v8f G = __builtin_amdgcn_wmma_f32_16x16x4_f32(
        false, ax, false, bx, (short)0, czero, false, false);
    v8f P = __builtin_amdgcn_wmma_f32_16x16x4_f32(
        false, am, false, bm, (short)0, czero, false, false);

    const float xxj = xjx * xjx + xjy * xjy + xjz * xjz;
    const float tj  = fmaf(-C1, xxj, 0.0f);

#pragma unroll
    for (int r = 0; r < 8; ++r) {
      // arg = log2e * -d2/s^2 = C1*(2G - xx_i - xx_j)
      float arg = fmaf(G[r], 2.0f * C1, negrow[r] + tj);
      float K   = __builtin_amdgcn_exp2f(arg);   // v_exp_f32
      float w   = P[r] * K;
      aKx[r] = fmaf(K, mjx, aKx[r]);
      aKy[r] = fmaf(K, mjy, aKy[r]);
      aKz[r] = fmaf(K, mjz, aKz[r]);
      aW[r] += w;
      aWx[r] = fmaf(w, xjx, aWx[r]);
      aWy[r] = fmaf(w, xjy, aWy[r]);
      aWz[r] = fmaf(w, xjz, aWz[r]);
    }
  }

  // ---- reduce across the 16 columns (stays inside each 16-lane half)
#pragma unroll
  for (int mask = 1; mask <= 8; mask <<= 1) {
#pragma unroll
    for (int r = 0; r < 8; ++r) {
      aW[r]  += __shfl_xor(aW[r],  mask, 32);
      aWx[r] += __shfl_xor(aWx[r], mask, 32);
      aWy[r] += __shfl_xor(aWy[r], mask, 32);
      aWz[r] += __shfl_xor(aWz[r], mask, 32);
      aKx[r] += __shfl_xor(aKx[r], mask, 32);
      aKy[r] += __shfl_xor(aKy[r], mask, 32);
      aKz[r] += __shfl_xor(aKz[r], mask, 32);
    }
  }

  // lane with col==r in each half owns row (r + 8*half); write wave slot
#pragma unroll
  for (int r = 0; r < 8; ++r) {
    if (col == r) {
      const int rl = r + 8 * half;
      red[wave][rl][0] = aW[r];
      red[wave][rl][1] = aWx[r];
      red[wave][rl][2] = aWy[r];
      red[wave][rl][3] = aWz[r];
      red[wave][rl][4] = aKx[r];
      red[wave][rl][5] = aKy[r];
      red[wave][rl][6] = aKz[r];
    }
  }
  __syncthreads();

  // deterministic combine across the 8 waves
  if (tid < 112) {
    const int row = tid / 7, q = tid % 7;
    float s = 0.0f;
#pragma unroll
    for (int w = 0; w < 8; ++w) s += red[w][row][q];
    red[0][row][q] = s;
  }
  __syncthreads();

  // final outputs: 16 rows x 3 comps each of (-g_pos, g_mom)
  if (tid < 48) {
    const int row = tid / 3, c = tid % 3;
    const int i = i0 + row;
    const float Sw  = red[0][row][0];
    const float Swx = red[0][row][1 + c];
    const float xi  = pos[3 * i + c];
    out[3 * i + c]         = POSCOEF * (Sw * xi - Swx);   // -g_pos
    out[3 * N + 3 * i + c] = 2.0f * red[0][row][4 + c];   // g_mom
  }
}

extern "C" void kernel_launch(void* const* d_in, const int* in_sizes, int n_in,
                              void* d_out, int out_size, void* d_ws, size_t ws_size,
                              hipStream_t stream) {
  const float* mom = (const float*)d_in[0];   // setup_inputs order: mom, pos
  const float* pos = (const float*)d_in[1];
  float* out = (float*)d_out;
  const int N = in_sizes[0] / 3;              // 8192
  const int tiles = (N + 15) / 16;            // 512 i-tiles, one block each
  LDDMMHamilton_wmma_kernel<<<dim3(tiles), dim3(256), 0, stream>>>(mom, pos, out, N);
}